// CausalUnitRunEncoder_85109071937710
// MI455X (gfx1250) — compile-verified
//
#include <hip/hip_runtime.h>
#include <math.h>

#define TC   64          // output timesteps per block
#define HALO 32          // recomputed halo (cumulative conv lookback = 30)
#define W    96          // TC + HALO
#define D    64
#define T_LEN 16384

typedef __attribute__((ext_vector_type(2))) float v2f;
typedef __attribute__((ext_vector_type(8))) float v8f;

__device__ __forceinline__ float clip01(float x) { return fminf(fmaxf(x, 0.f), 1.f); }
__device__ __forceinline__ float gelu_exact(float x) {
  return 0.5f * x * (1.f + erff(x * 0.70710678118654752440f));
}

__global__ __launch_bounds__(256) void causal_unit_run_encoder(
    const int*   __restrict__ unit_ids,
    const float* __restrict__ log_anchor,
    const float* __restrict__ source_rate,
    const float* __restrict__ silence_mask,
    const float* __restrict__ unit_mask,
    const float* __restrict__ emb,     // [2048,64]
    const float* __restrict__ in_w,    // [64,67]
    const float* __restrict__ in_b,    // [64]
    const float* __restrict__ dw_w,    // [4,64,1,3]
    const float* __restrict__ dw_b,    // [4,64]
    const float* __restrict__ pw_w,    // [4,64,64]
    const float* __restrict__ pw_b,    // [4,64]
    const float* __restrict__ ln_g,    // [64]
    const float* __restrict__ ln_b,    // [64]
    float*       __restrict__ out)     // [B,T,64]
{
  __shared__ __align__(16) float sh_h[W * D];    // residual stream, 24 KB
  __shared__ __align__(16) float sh_a[W * 68];   // features (stride 68) / conv act (stride 64), 26 KB
  __shared__ float sh_mask[W];

  const int    tid     = threadIdx.x;
  const int    b       = blockIdx.y;
  const int    t0w     = blockIdx.x * TC - HALO; // global t of window row 0 (may be <0)
  const size_t base_bt = (size_t)b * T_LEN;

  // ---- Stage 0a: gather embeddings + scalar features into sh_a [W][68] ----
  for (int task = tid; task < W * 17; task += 256) {
    int r  = task / 17;
    int ch = task - r * 17;
    int t  = t0w + r;
    if (ch == 16) {
      float la = 0.f, ce = 0.f, si = 0.f, mk = 0.f;
      if (t >= 0) {
        size_t off = base_bt + t;
        mk = clip01(unit_mask[off]);
        la = log_anchor[off];
        ce = (la - source_rate[off]) * mk;
        si = clip01(silence_mask[off]) * mk;
      }
      *(float4*)&sh_a[r * 68 + 64] = make_float4(la, ce, si, 0.f);
      sh_mask[r] = mk;
    } else {
      float4 v = make_float4(0.f, 0.f, 0.f, 0.f);
      if (t >= 0) {
        int id = unit_ids[base_bt + t];
        v = *(const float4*)&emb[(size_t)id * D + ch * 4];
      }
      *(float4*)&sh_a[r * 68 + ch * 4] = v;
    }
  }
  __syncthreads();

  // WMMA fragment lane mapping (f32 16x16x4, per CDNA5 ISA 7.12.2):
  //  A (16x4): lane%16 = M row; VGPR0 holds K=khalf, VGPR1 K=khalf+1 (khalf=0|2)
  //  B (4x16): lane%16 = N col; VGPR0 holds K=khalf, VGPR1 K=khalf+1
  //  C/D:      VGPR r -> M = r + (lane>=16 ? 8:0), N = lane%16
  const int wid   = tid >> 5;
  const int lane  = tid & 31;
  const int n     = lane & 15;
  const int khalf = (lane >> 4) << 1;   // 0 or 2
  const int d0    = (wid & 3) * 16;     // d-tile per wave
  const int mrow  = lane & 15;
  const int macc  = (lane >> 4) << 3;   // 0 or 8

  // ---- Stage 0b: input projection  h[t,d] = (sum_f feat[t,f]*in_w[d,f] + in_b[d]) * mask[t] ----
  {
    v2f bw[17];
#pragma unroll
    for (int kk = 0; kk < 17; ++kk) {
      int f0 = kk * 4 + khalf;
      int d  = d0 + n;
      bw[kk].x = (f0     < 67) ? in_w[d * 67 + f0]     : 0.f;
      bw[kk].y = (f0 + 1 < 67) ? in_w[d * 67 + f0 + 1] : 0.f;
    }
    float bias = in_b[d0 + n];
    for (int tt = (wid >> 2); tt < W / 16; tt += 2) {
      const float* ap = &sh_a[(tt * 16 + mrow) * 68 + khalf];
      v8f acc = {};
#pragma unroll
      for (int kk = 0; kk < 17; ++kk) {
        v2f a;
        a.x = ap[kk * 4];
        a.y = ap[kk * 4 + 1];
        acc = __builtin_amdgcn_wmma_f32_16x16x4_f32(false, a, false, bw[kk],
                                                    (short)0, acc, false, false);
      }
#pragma unroll
      for (int r = 0; r < 8; ++r) {
        int row = tt * 16 + macc + r;
        sh_h[row * D + d0 + n] = (acc[r] + bias) * sh_mask[row];
      }
    }
  }
  __syncthreads();

  // ---- 4 residual blocks: depthwise causal dilated conv -> GELU -> pointwise -> add ----
  const int dil_arr[4] = {1, 2, 4, 8};
  for (int s = 0; s < 4; ++s) {
    const int dil = dil_arr[s];

    // depthwise conv (taps t, t-dil, t-2*dil with w[2],w[1],w[0]) + bias + exact GELU -> sh_a [W][64]
    for (int idx = tid; idx < W * D; idx += 256) {
      int r = idx >> 6;
      int d = idx & 63;
      const float* wv = &dw_w[s * 192 + d * 3];
      float x = wv[2] * sh_h[r * D + d];
      int r1 = r - dil;
      if (r1 >= 0) x += wv[1] * sh_h[r1 * D + d];
      int r2 = r - 2 * dil;
      if (r2 >= 0) x += wv[0] * sh_h[r2 * D + d];
      x += dw_b[s * D + d];
      sh_a[idx] = gelu_exact(x);
    }
    __syncthreads();

    // pointwise: h[t,o] += sum_c u[t,c]*pw_w[s,o,c] + pw_b[s,o]   (WMMA f32 16x16x4, K=64)
    {
      const float* pw = &pw_w[s * D * D];
      v2f bw[16];
#pragma unroll
      for (int kk = 0; kk < 16; ++kk) {
        int c0 = kk * 4 + khalf;
        bw[kk].x = pw[(d0 + n) * D + c0];
        bw[kk].y = pw[(d0 + n) * D + c0 + 1];
      }
      float bias = pw_b[s * D + d0 + n];
      for (int tt = (wid >> 2); tt < W / 16; tt += 2) {
        const float* ap = &sh_a[(tt * 16 + mrow) * D + khalf];
        v8f acc = {};
#pragma unroll
        for (int kk = 0; kk < 16; ++kk) {
          v2f a;
          a.x = ap[kk * 4];
          a.y = ap[kk * 4 + 1];
          acc = __builtin_amdgcn_wmma_f32_16x16x4_f32(false, a, false, bw[kk],
                                                      (short)0, acc, false, false);
        }
#pragma unroll
        for (int r = 0; r < 8; ++r) {
          int row = tt * 16 + macc + r;
          sh_h[row * D + d0 + n] += acc[r] + bias;
        }
      }
    }
    __syncthreads();
  }

  // ---- LayerNorm over D, scale/shift, mask, write valid rows [HALO, W) ----
  if (tid < TC) {
    int r = HALO + tid;
    int t = blockIdx.x * TC + tid;
    const float* hp = &sh_h[r * D];
    float mu = 0.f;
#pragma unroll
    for (int d = 0; d < D; ++d) mu += hp[d];
    mu *= (1.f / D);
    float var = 0.f;
#pragma unroll
    for (int d = 0; d < D; ++d) { float c = hp[d] - mu; var += c * c; }
    var *= (1.f / D);
    float rs = 1.0f / sqrtf(var + 1e-5f);
    float mk = sh_mask[r];
    float* op = &out[(base_bt + t) * D];
#pragma unroll
    for (int d = 0; d < D; d += 4) {
      float4 v;
      v.x = ((hp[d + 0] - mu) * rs * ln_g[d + 0] + ln_b[d + 0]) * mk;
      v.y = ((hp[d + 1] - mu) * rs * ln_g[d + 1] + ln_b[d + 1]) * mk;
      v.z = ((hp[d + 2] - mu) * rs * ln_g[d + 2] + ln_b[d + 2]) * mk;
      v.w = ((hp[d + 3] - mu) * rs * ln_g[d + 3] + ln_b[d + 3]) * mk;
      *(float4*)&op[d] = v;
    }
  }
}

extern "C" void kernel_launch(void* const* d_in, const int* in_sizes, int n_in,
                              void* d_out, int out_size, void* d_ws, size_t ws_size,
                              hipStream_t stream) {
  (void)in_sizes; (void)n_in; (void)out_size; (void)d_ws; (void)ws_size;
  dim3 grid(T_LEN / TC, 32);
  causal_unit_run_encoder<<<grid, 256, 0, stream>>>(
      (const int*)d_in[0],   (const float*)d_in[1], (const float*)d_in[2],
      (const float*)d_in[3], (const float*)d_in[4], (const float*)d_in[5],
      (const float*)d_in[6], (const float*)d_in[7], (const float*)d_in[8],
      (const float*)d_in[9], (const float*)d_in[10], (const float*)d_in[11],
      (const float*)d_in[12], (const float*)d_in[13], (float*)d_out);
}